// ROPE_23476291240446
// MI455X (gfx1250) — compile-verified
//
#include <hip/hip_runtime.h>

// ---------------------------------------------------------------------------
// RoPE, fp32, shape [4,32,4096,128]. Pure HBM-streaming kernel:
//   537 MB total traffic / 23.3 TB/s  =>  ~23 us roofline floor.
// Strategy: CDNA5 async global->LDS B128 pipeline (ASYNCcnt) 8 deep for
// ~32 MB aggregate in-flight traffic, NT 128-bit stores, hoisted per-lane
// RoPE frequencies, hardware v_sin/v_cos with v_fract range reduction.
// ---------------------------------------------------------------------------

typedef float v4f __attribute__((ext_vector_type(4)));
typedef int   v4i __attribute__((ext_vector_type(4)));

#define SEQ_MASK 4095   // S = 4096
#define COL_MASK 31     // D/2/2 = 32 float4 columns per row (D = 128)

// log2(10000)/64 : freqs_i = 2^(-i*K), i in [0,64)
#define ROPE_K    0.20762050593045702f
#define INV_2PI   0.15915494309189535f

#if __has_builtin(__builtin_amdgcn_exp2f)
__device__ __forceinline__ float fast_exp2(float x) { return __builtin_amdgcn_exp2f(x); }
#else
__device__ __forceinline__ float fast_exp2(float x) { return exp2f(x); }
#endif

#if __has_builtin(__builtin_amdgcn_fractf)
__device__ __forceinline__ float fast_fract(float x) { return __builtin_amdgcn_fractf(x); }
#else
__device__ __forceinline__ float fast_fract(float x) { return x - floorf(x); }
#endif

#if __has_builtin(__builtin_amdgcn_sinf) && __has_builtin(__builtin_amdgcn_cosf)
// Hardware v_sin/v_cos take angle in turns (pre-scaled by 1/2pi).
__device__ __forceinline__ float fast_sin_turns(float t) { return __builtin_amdgcn_sinf(t); }
__device__ __forceinline__ float fast_cos_turns(float t) { return __builtin_amdgcn_cosf(t); }
#else
__device__ __forceinline__ float fast_sin_turns(float t) { return __sinf(t * 6.2831853071795864f); }
__device__ __forceinline__ float fast_cos_turns(float t) { return __cosf(t * 6.2831853071795864f); }
#endif

// Rotate one float4 = rotation pairs (x,y) @ freq f0 and (z,w) @ freq f1.
// f0/f1 are pre-scaled by 1/(2pi) so fract() is the exact range reduction.
__device__ __forceinline__ v4f rope_quad(v4f v, float sf, float f0, float f1) {
    float t0 = fast_fract(sf * f0);
    float t1 = fast_fract(sf * f1);
    float s0 = fast_sin_turns(t0), c0 = fast_cos_turns(t0);
    float s1 = fast_sin_turns(t1), c1 = fast_cos_turns(t1);
    v4f o;
    o.x = fmaf(-v.y, s0, v.x * c0);
    o.y = fmaf( v.x, s0, v.y * c0);
    o.z = fmaf(-v.w, s1, v.z * c1);
    o.w = fmaf( v.z, s1, v.w * c1);
    return o;
}

// ---------------------------------------------------------------------------
// CDNA5 path: async global->LDS B128 pipeline (ASYNCcnt), DEPTH=8 stages.
// Each lane fills and drains its own LDS slot -> no barriers, only
// s_wait_asynccnt (+ s_wait_dscnt before slot reuse). NT stores from VGPRs.
// ---------------------------------------------------------------------------
#define HAVE_GFX1250_ASYNC \
    (__has_builtin(__builtin_amdgcn_global_load_async_to_lds_b128) && \
     __has_builtin(__builtin_amdgcn_s_wait_asynccnt))

#if HAVE_GFX1250_ASYNC

#define TPB    256
#define TILES  16
#define DEPTH  8

#define LOCAL_AS  __attribute__((address_space(3)))

__global__ __launch_bounds__(TPB) void rope_async(const v4f* __restrict__ x,
                                                  v4f* __restrict__ y) {
    __shared__ v4f buf[DEPTH][TPB];   // 32 KB
    const int tid = threadIdx.x;

    // Tile bases are multiples of 256 -> column index is tile-invariant.
    const int c = tid & COL_MASK;
    const float i0 = (float)(2 * c);
    const float f0 = fast_exp2(-i0 * ROPE_K) * INV_2PI;
    const float f1 = fast_exp2(-(i0 + 1.0f) * ROPE_K) * INV_2PI;

    const long long blk = blockIdx.x;
    const long long gstride = gridDim.x;

    // Prime the pipeline: DEPTH async B128 loads in flight per wave.
    #pragma unroll
    for (int j = 0; j < DEPTH; ++j) {
        long long g = (blk + (long long)j * gstride) * TPB + tid;
        __builtin_amdgcn_global_load_async_to_lds_b128(
            (v4i*)(x + g),
            (LOCAL_AS v4i*)(&buf[j][tid]),
            0, 0);
    }

    // Steady state: oldest stage guaranteed complete once ASYNCcnt <= DEPTH-1.
    for (int j = 0; j < TILES - DEPTH; ++j) {
        __builtin_amdgcn_s_wait_asynccnt(DEPTH - 1);
        __asm__ volatile("" ::: "memory");
        long long g = (blk + (long long)j * gstride) * TPB + tid;
        int s = (int)((g >> 5) & SEQ_MASK);
        v4f v = buf[j & (DEPTH - 1)][tid];
        v4f o = rope_quad(v, (float)s, f0, f1);
        __builtin_nontemporal_store(o, y + g);
        // LDS read must have landed in VGPRs before the async engine may
        // overwrite the slot (DS vs async ops are unordered).
        __asm__ volatile("s_wait_dscnt 0" ::: "memory");
        long long gn = (blk + (long long)(j + DEPTH) * gstride) * TPB + tid;
        __builtin_amdgcn_global_load_async_to_lds_b128(
            (v4i*)(x + gn),
            (LOCAL_AS v4i*)(&buf[j & (DEPTH - 1)][tid]),
            0, 0);
    }

    // Drain: everything outstanding has landed after asynccnt==0.
    __builtin_amdgcn_s_wait_asynccnt(0);
    __asm__ volatile("" ::: "memory");
    #pragma unroll
    for (int j = TILES - DEPTH; j < TILES; ++j) {
        long long g = (blk + (long long)j * gstride) * TPB + tid;
        int s = (int)((g >> 5) & SEQ_MASK);
        v4f v = buf[j & (DEPTH - 1)][tid];
        v4f o = rope_quad(v, (float)s, f0, f1);
        __builtin_nontemporal_store(o, y + g);
    }
}
#endif  // HAVE_GFX1250_ASYNC

// ---------------------------------------------------------------------------
// Fallback: plain vectorized streaming kernel (one float4 per thread).
// ---------------------------------------------------------------------------
__global__ __launch_bounds__(256) void rope_simple(const v4f* __restrict__ x,
                                                   v4f* __restrict__ y, int n4) {
    int g = blockIdx.x * blockDim.x + threadIdx.x;
    if (g >= n4) return;
    int c = g & COL_MASK;
    int s = (g >> 5) & SEQ_MASK;
    float i0 = (float)(2 * c);
    float f0 = fast_exp2(-i0 * ROPE_K) * INV_2PI;
    float f1 = fast_exp2(-(i0 + 1.0f) * ROPE_K) * INV_2PI;
    v4f v = __builtin_nontemporal_load(x + g);
    v4f o = rope_quad(v, (float)s, f0, f1);
    __builtin_nontemporal_store(o, y + g);
}

// ---------------------------------------------------------------------------
extern "C" void kernel_launch(void* const* d_in, const int* in_sizes, int n_in,
                              void* d_out, int out_size, void* d_ws, size_t ws_size,
                              hipStream_t stream) {
    const float* x = (const float*)d_in[0];
    float* y = (float*)d_out;
    long long n = in_sizes[0];      // total fp32 elements: 4*32*4096*128
    long long n4 = n >> 2;          // float4 elements: 16,777,216

#if HAVE_GFX1250_ASYNC
    const long long per_block = (long long)TPB * TILES;   // 4096 float4 / block
    if (n4 % per_block == 0) {
        int grid = (int)(n4 / per_block);                 // 4096 blocks
        rope_async<<<grid, TPB, 0, stream>>>((const v4f*)x, (v4f*)y);
        return;
    }
#endif
    int grid = (int)((n4 + 255) / 256);
    rope_simple<<<grid, 256, 0, stream>>>((const v4f*)x, (v4f*)y, (int)n4);
}